// AdaptiveDenoising_67860483277346
// MI455X (gfx1250) — compile-verified
//
#include <hip/hip_runtime.h>
#include <stdint.h>

// Bilateral blur 5x5, sigma_s=1.0, sigma_r=0.06, NHWC f32, reflect padding.
// CDNA5 path: per-lane async global->LDS halo loads (reflect folded into the
// per-lane address math, one addr per 12B pixel reused via offset:0/4/8),
// s_wait_asynccnt, LDS-resident 25-tap stencil.

#define KS      5
#define PAD     2
#define TILE_W  64
#define TILE_H  16
#define HALO_W  (TILE_W + 2 * PAD)      // 68 pixels
#define HALO_H  (TILE_H + 2 * PAD)      // 20 rows
#define HALO_PIX (HALO_W * HALO_H)      // 1360 pixels
#define ROW_F   (HALO_W * 3)            // 204 floats per halo row (816 B, 16B-aligned)
#define TILE_FLOATS (HALO_H * ROW_F)    // 4080 floats = 16320 B LDS

#define IMG_H 1536
#define IMG_W 2048
#define IMG_C 3

__device__ __forceinline__ int reflect_i(int i, int n) {
    // jnp.pad mode="reflect" for i in [-2, n+1], branchless:
    // min( max(i, -i), 2n-2-i )
    const int a = (i < 0) ? -i : i;
    const int b = 2 * n - 2 - i;
    return (a < b) ? a : b;
}

__global__ __launch_bounds__(256)
void bilateral5x5_kernel(const float* __restrict__ img, float* __restrict__ out) {
    __shared__ float tile[TILE_FLOATS];

    const int tx  = threadIdx.x;            // 0..15
    const int ty  = threadIdx.y;            // 0..15
    const int tid = ty * 16 + tx;           // 0..255

    const int bx0 = blockIdx.x * TILE_W;
    const int by0 = blockIdx.y * TILE_H;
    const int b   = blockIdx.z;

    const uint64_t srcBase =
        (uint64_t)(uintptr_t)(img + (size_t)b * IMG_H * IMG_W * IMG_C);

    // ------- async halo load: global -> LDS (reflect padded) -----------------
    // 1360 pixels / 256 threads = 6 steps (last guarded). One address pair per
    // 12B pixel; offset:0/4/8 reuses it for the 3 channels (INST_OFFSET is
    // added to both the LDS and the global address). Index math is one
    // div-by-68 at entry, then increments: j += 256 = 3*68 + 52.
    int r  = tid / HALO_W;                  // halo row 0..3
    int px = tid - r * HALO_W;              // halo pixel 0..67
    uint32_t ldsPix = (uint32_t)(uintptr_t)tile + (uint32_t)tid * 12u;
#pragma unroll
    for (int k = 0; k < 6; ++k) {
        if (k < 5 || tid < (HALO_PIX - 5 * 256)) {   // last step: 80 pixels
            const int gy = reflect_i(by0 + r  - PAD, IMG_H);
            const int gx = reflect_i(bx0 + px - PAD, IMG_W);
            const uint32_t gOff = (uint32_t)(gy * (IMG_W * 12) + gx * 12);
            asm volatile("global_load_async_to_lds_b32 %0, %1, %2"
                         :: "v"(ldsPix), "v"(gOff), "s"(srcBase) : "memory");
            asm volatile("global_load_async_to_lds_b32 %0, %1, %2 offset:4"
                         :: "v"(ldsPix), "v"(gOff), "s"(srcBase) : "memory");
            asm volatile("global_load_async_to_lds_b32 %0, %1, %2 offset:8"
                         :: "v"(ldsPix), "v"(gOff), "s"(srcBase) : "memory");
        }
        px += 256 - 3 * HALO_W;             // +52
        const int wrap = (px >= HALO_W) ? 1 : 0;
        px -= wrap ? HALO_W : 0;
        r  += 3 + wrap;
        ldsPix += 256u * 12u;               // LDS addr is linear in pixel index
    }
    asm volatile("s_wait_asynccnt 0x0" ::: "memory");
    __syncthreads();

    // ---------------- 25-tap bilateral from LDS ------------------------------
    const int lx0 = tx * 4 + PAD;   // first output pixel (halo coords)
    const int ly  = ty + PAD;

    float c[4][3];                  // centers, extracted from the dy=2 row regs
    float num[4][3] = {};
    float den[4]    = {};

    // log2 of normalized 1D gaussian (ks=5, sigma=1.0)
    const float L2G[5] = {-4.1978994f, -2.0338568f, -1.3125095f,
                          -2.0338568f, -4.1978994f};
    // (-0.5 / 0.06^2) * log2(e): range kernel exponent scale
    constexpr float K2 = -200.37431956791158f;

    // process the center row first so centers come from its row registers
    constexpr int DY[5] = {2, 0, 1, 3, 4};
#pragma unroll
    for (int q = 0; q < 5; ++q) {
        const int dy  = DY[q];
        const int row = ly + dy - PAD;
        // 8 neighbor pixels (24 floats) covering all 4 strips x 5 dx
        float rv[24];
#pragma unroll
        for (int j = 0; j < 24; ++j)
            rv[j] = tile[row * ROW_F + (lx0 - PAD) * 3 + j];

        if (q == 0) {               // dy == 2: centers are rv[(p+2)*3 + cc]
#pragma unroll
            for (int p = 0; p < 4; ++p)
#pragma unroll
                for (int cc = 0; cc < 3; ++cc)
                    c[p][cc] = rv[(p + 2) * 3 + cc];
        }

#pragma unroll
        for (int dx = 0; dx < 5; ++dx) {
            const float l2sk = L2G[dy] + L2G[dx];   // compile-time constant
#pragma unroll
            for (int p = 0; p < 4; ++p) {
                const float n0 = rv[(p + dx) * 3 + 0];
                const float n1 = rv[(p + dx) * 3 + 1];
                const float n2 = rv[(p + dx) * 3 + 2];
                float cd = fabsf(n0 - c[p][0]) + fabsf(n1 - c[p][1]) +
                           fabsf(n2 - c[p][2]);
                cd = cd * cd;                       // (sum |diff|)^2
                // w = space_k * exp(inv2sr2*cd) = exp2(fma(cd, K2, log2(space_k)))
                const float w = __builtin_amdgcn_exp2f(fmaf(cd, K2, l2sk));
                num[p][0] = fmaf(n0, w, num[p][0]);
                num[p][1] = fmaf(n1, w, num[p][1]);
                num[p][2] = fmaf(n2, w, num[p][2]);
                den[p] += w;
            }
        }
    }

    // ---------------- normalize, clip, store (3 x b128) ----------------------
    float ov[12];
#pragma unroll
    for (int p = 0; p < 4; ++p) {
        const float rcp = __builtin_amdgcn_rcpf(den[p]);
#pragma unroll
        for (int cc = 0; cc < 3; ++cc) {
            float v = num[p][cc] * rcp;
            v = fminf(fmaxf(v, 0.0f), 1.0f);
            ov[p * 3 + cc] = v;
        }
    }

    const int gy  = by0 + ty;
    const int gx0 = bx0 + tx * 4;
    float* op = out + (((size_t)b * IMG_H + gy) * IMG_W + gx0) * 3; // 48B-aligned
    float4* o4 = reinterpret_cast<float4*>(op);
    o4[0] = make_float4(ov[0], ov[1], ov[2],  ov[3]);
    o4[1] = make_float4(ov[4], ov[5], ov[6],  ov[7]);
    o4[2] = make_float4(ov[8], ov[9], ov[10], ov[11]);
}

extern "C" void kernel_launch(void* const* d_in, const int* in_sizes, int n_in,
                              void* d_out, int out_size, void* d_ws, size_t ws_size,
                              hipStream_t stream) {
    const float* img = (const float*)d_in[0];
    float* out = (float*)d_out;
    const int B = in_sizes[0] / (IMG_H * IMG_W * IMG_C);   // = 8
    dim3 block(16, 16, 1);
    dim3 grid(IMG_W / TILE_W, IMG_H / TILE_H, B);          // (32, 96, 8)
    bilateral5x5_kernel<<<grid, block, 0, stream>>>(img, out);
}